// GATModel_75788992905473
// MI455X (gfx1250) — compile-verified
//
#include <hip/hip_runtime.h>

// ---------------------------------------------------------------------------
// GAT (2-layer, heads=2, concat=False) for MI455X / gfx1250, fp32 throughout.
// GEMMs use V_WMMA_F32_16X16X4_F32; edge phase uses L2-resident atomics.
// ---------------------------------------------------------------------------

#define GAT_H 2
#define NEG_SLOPE 0.2f

typedef float v2f __attribute__((ext_vector_type(2)));
typedef float v8f __attribute__((ext_vector_type(8)));

// ---- monotonic float<->uint encoding for atomicMax-based segment max ------
__device__ __forceinline__ unsigned encodeF(float x) {
  unsigned u = __float_as_uint(x);
  return (u & 0x80000000u) ? ~u : (u | 0x80000000u);
}
__device__ __forceinline__ float decodeF(unsigned u) {
  return __uint_as_float((u & 0x80000000u) ? (u ^ 0x80000000u) : ~u);
}

// ---------------------------------------------------------------------------
// Utility zero / fill kernels (accumulators must be re-zeroed every call)
// ---------------------------------------------------------------------------
__global__ void k_zero_f32(float* __restrict__ p, long n) {
  long i = (long)blockIdx.x * blockDim.x + threadIdx.x;
  if (i < n) p[i] = 0.0f;
}
__global__ void k_fill_u32(unsigned* __restrict__ p, long n, unsigned v) {
  long i = (long)blockIdx.x * blockDim.x + threadIdx.x;
  if (i < n) p[i] = v;
}

// ---------------------------------------------------------------------------
// Self-loop attr = mean of incoming edge attrs (segment mean over dst)
// ---------------------------------------------------------------------------
__global__ void k_loop_accum(const int* __restrict__ ei, const float* __restrict__ ea,
                             float* __restrict__ ssum, float* __restrict__ cnt, int E) {
  int e = blockIdx.x * blockDim.x + threadIdx.x;
  if (e >= E) return;
  int d = ei[E + e];                       // edge_index[1][e]
  atomicAdd(&ssum[d * 2 + 0], ea[e * 2 + 0]);
  atomicAdd(&ssum[d * 2 + 1], ea[e * 2 + 1]);
  atomicAdd(&cnt[d], 1.0f);
}

__global__ void k_build_ext(const int* __restrict__ ei, const float* __restrict__ ea,
                            const float* __restrict__ ssum, const float* __restrict__ cnt,
                            int* __restrict__ srcX, int* __restrict__ dstX,
                            float* __restrict__ eaX, int E, int N) {
  int e = blockIdx.x * blockDim.x + threadIdx.x;
  int EX = E + N;
  if (e >= EX) return;
  if (e < E) {
    srcX[e] = ei[e];
    dstX[e] = ei[E + e];
    eaX[e * 2 + 0] = ea[e * 2 + 0];
    eaX[e * 2 + 1] = ea[e * 2 + 1];
  } else {
    int n = e - E;
    srcX[e] = n;
    dstX[e] = n;
    float c = fmaxf(cnt[n], 1.0f);
    eaX[e * 2 + 0] = ssum[n * 2 + 0] / c;
    eaX[e * 2 + 1] = ssum[n * 2 + 1] / c;
  }
}

// ---------------------------------------------------------------------------
// fp32 WMMA GEMM: C[M,Nc] = A[M,K] * B[K,Nc], row-major.
// One block = 16 rows of A (in LDS) x full Nc; 8 waves, each wave owns a
// 16x16 output tile, K-loop in steps of 4 via v_wmma_f32_16x16x4_f32.
// A frag (16x4 f32): lanes 0-15 rows, VGPR0/1 = K{0,1} (lo half) / K{2,3} (hi).
// B frag (4x16 f32): lanes hold N; VGPR0/1 = K{0,1} (lo half) / K{2,3} (hi).
// C frag: VGPR v = row v (lanes 0-15) / row 8+v (lanes 16-31), col = lane%16.
// ---------------------------------------------------------------------------
__global__ __launch_bounds__(256) void k_gemm_wmma_f32(
    const float* __restrict__ A, const float* __restrict__ B,
    float* __restrict__ C, int M, int K, int Nc) {
  extern __shared__ float ldsA[];          // 16 * K floats
  const int rowBase = blockIdx.x * 16;
  const int tid = threadIdx.x;

  for (int i = tid; i < 16 * K; i += blockDim.x) {
    int r = i / K, k = i - r * K;
    int gr = rowBase + r;
    ldsA[i] = (gr < M) ? A[(long)gr * K + k] : 0.0f;
  }
  __syncthreads();

  const int wave   = tid >> 5;
  const int lane   = tid & 31;
  const int colBase = wave * 16;
  const int laneHi  = lane >> 4;           // 0: lanes 0-15, 1: lanes 16-31
  const int lane16  = lane & 15;

  v8f c = {};
  for (int k0 = 0; k0 < K; k0 += 4) {
    const int ka = k0 + laneHi * 2;
    v2f a, b;
    a.x = ldsA[lane16 * K + ka + 0];
    a.y = ldsA[lane16 * K + ka + 1];
    b.x = B[(long)(ka + 0) * Nc + colBase + lane16];
    b.y = B[(long)(ka + 1) * Nc + colBase + lane16];
    c = __builtin_amdgcn_wmma_f32_16x16x4_f32(
        /*neg_a=*/false, a, /*neg_b=*/false, b,
        /*c_mod=*/(short)0, c, /*reuse_a=*/false, /*reuse_b=*/false);
  }

  const int rOff = laneHi * 8;
#pragma unroll
  for (int v = 0; v < 8; ++v) {
    int gr = rowBase + rOff + v;
    if (gr < M) C[(long)gr * Nc + colBase + lane16] = c[v];
  }
}

// ---------------------------------------------------------------------------
// Per-node attention logits: a_src[n,h], a_dst[n,h]
// ---------------------------------------------------------------------------
__global__ void k_node_att(const float* __restrict__ Hf,
                           const float* __restrict__ att_src,
                           const float* __restrict__ att_dst,
                           float* __restrict__ a_src, float* __restrict__ a_dst,
                           int N, int Cc) {
  int i = blockIdx.x * blockDim.x + threadIdx.x;
  if (i >= N * GAT_H) return;
  int n = i >> 1, h = i & 1;
  const float* row = Hf + (long)n * (GAT_H * Cc) + h * Cc;
  const float* as = att_src + h * Cc;
  const float* ad = att_dst + h * Cc;
  float s = 0.0f, d = 0.0f;
  for (int c = 0; c < Cc; ++c) {
    float f = row[c];
    s += f * as[c];
    d += f * ad[c];
  }
  a_src[i] = s;
  a_dst[i] = d;
}

// we_att[j*2+h] = sum_c W_edge[j, h*C+c] * att_edge[h, c]   (2x2 matrix)
__global__ void k_we_att(const float* __restrict__ W_edge,
                         const float* __restrict__ att_edge,
                         float* __restrict__ weatt, int Cc) {
  int i = threadIdx.x;
  if (i >= 4) return;
  int j = i >> 1, h = i & 1;
  float s = 0.0f;
  for (int c = 0; c < Cc; ++c)
    s += W_edge[j * (GAT_H * Cc) + h * Cc + c] * att_edge[h * Cc + c];
  weatt[i] = s;
}

// ---------------------------------------------------------------------------
// Edge pass 1: raw logits + leaky relu + segment max (atomicMax on encoded u32)
// ---------------------------------------------------------------------------
__global__ void k_edge_alpha(const int* __restrict__ srcX, const int* __restrict__ dstX,
                             const float* __restrict__ eaX,
                             const float* __restrict__ a_src, const float* __restrict__ a_dst,
                             const float* __restrict__ weatt,
                             float* __restrict__ alphaE, unsigned* __restrict__ amax, long EX) {
  long i = (long)blockIdx.x * blockDim.x + threadIdx.x;
  if (i >= EX * 2) return;
  long e = i >> 1;
  int h = (int)(i & 1);
  int d = dstX[e];
  float al = a_src[srcX[e] * 2 + h] + a_dst[d * 2 + h] +
             eaX[e * 2 + 0] * weatt[0 * 2 + h] +
             eaX[e * 2 + 1] * weatt[1 * 2 + h];
  al = (al > 0.0f) ? al : NEG_SLOPE * al;
  alphaE[i] = al;
  atomicMax(&amax[d * 2 + h], encodeF(al));
}

// Edge pass 2: exp(alpha - max) in place + segment sum
__global__ void k_edge_exp(const int* __restrict__ dstX, float* __restrict__ alphaE,
                           const unsigned* __restrict__ amax, float* __restrict__ asum,
                           long EX) {
  long i = (long)blockIdx.x * blockDim.x + threadIdx.x;
  if (i >= EX * 2) return;
  long e = i >> 1;
  int h = (int)(i & 1);
  int d = dstX[e];
  float ex = expf(alphaE[i] - decodeF(amax[d * 2 + h]));
  alphaE[i] = ex;
  atomicAdd(&asum[d * 2 + h], ex);
}

// Edge pass 3: normalize (write output alpha) + float4 gather + atomic scatter
// 32 threads per edge, each handles 4 channels of the 128 (H*C) feature row.
__global__ void k_edge_scatter(const int* __restrict__ srcX, const int* __restrict__ dstX,
                               const float* __restrict__ alphaE, const float* __restrict__ asum,
                               const float* __restrict__ Hf, float* __restrict__ acc,
                               float* __restrict__ outAlpha, long EX, int HC) {
  const int Q = HC / 4;                        // 32 float4 chunks per edge
  long i = (long)blockIdx.x * blockDim.x + threadIdx.x;
  if (i >= EX * Q) return;
  long e = i / Q;
  int q = (int)(i - e * Q);
  int d = dstX[e], s = srcX[e];
  int h = (q * 4) / (HC / 2);                  // head for this chunk
  float a = alphaE[e * 2 + h] / (asum[d * 2 + h] + 1e-16f);
  if (q == 0) {
    outAlpha[e * 2 + 0] = alphaE[e * 2 + 0] / (asum[d * 2 + 0] + 1e-16f);
    outAlpha[e * 2 + 1] = alphaE[e * 2 + 1] / (asum[d * 2 + 1] + 1e-16f);
  }
  const float4 f = ((const float4*)(Hf + (long)s * HC))[q];
  float* ap = acc + (long)d * HC + q * 4;
  atomicAdd(ap + 0, f.x * a);
  atomicAdd(ap + 1, f.y * a);
  atomicAdd(ap + 2, f.z * a);
  atomicAdd(ap + 3, f.w * a);
}

// Head mean + bias (+ optional ReLU)
__global__ void k_finalize(const float* __restrict__ acc, const float* __restrict__ bias,
                           float* __restrict__ out, int N, int Cc, int doRelu) {
  long i = (long)blockIdx.x * blockDim.x + threadIdx.x;
  if (i >= (long)N * Cc) return;
  int n = (int)(i / Cc), c = (int)(i - (long)n * Cc);
  float v = (acc[(long)n * 2 * Cc + c] + acc[(long)n * 2 * Cc + Cc + c]) * 0.5f + bias[c];
  if (doRelu) v = fmaxf(v, 0.0f);
  out[i] = v;
}

// ---------------------------------------------------------------------------
// Host side
// ---------------------------------------------------------------------------
static inline int nblk(long n, int b) { return (int)((n + b - 1) / b); }

struct LayerBufs {
  const int *srcX, *dstX;
  const float* eaX;
  float *Hfeat, *a_src, *a_dst, *asum, *alphaE, *acc, *weatt;
  unsigned* amax;
};

static void run_gat_layer(const float* Xin, int K, const float* W,
                          const float* att_s, const float* att_d,
                          const float* W_edge, const float* att_e, const float* bias,
                          float* outAlpha, float* outFeat, int doRelu,
                          int N, long EX, const LayerBufs& b, hipStream_t stream) {
  const int Cc = 64, HC = GAT_H * Cc;          // 128 output columns
  // 1) Hfeat = Xin @ W  (WMMA)
  k_gemm_wmma_f32<<<nblk(N, 16), 256, 16 * K * (int)sizeof(float), stream>>>(
      Xin, W, b.Hfeat, N, K, HC);
  // 2) per-node attention dots + 2x2 edge-attention matrix
  k_node_att<<<nblk((long)N * GAT_H, 256), 256, 0, stream>>>(
      b.Hfeat, att_s, att_d, b.a_src, b.a_dst, N, Cc);
  k_we_att<<<1, 32, 0, stream>>>(W_edge, att_e, b.weatt, Cc);
  // 3) reset segment-reduction buffers
  k_fill_u32<<<nblk((long)N * GAT_H, 256), 256, 0, stream>>>(b.amax, (long)N * GAT_H, 0u);
  k_zero_f32<<<nblk((long)N * GAT_H, 256), 256, 0, stream>>>(b.asum, (long)N * GAT_H);
  k_zero_f32<<<nblk((long)N * HC, 256), 256, 0, stream>>>(b.acc, (long)N * HC);
  // 4) edge passes
  k_edge_alpha<<<nblk(EX * 2, 256), 256, 0, stream>>>(
      b.srcX, b.dstX, b.eaX, b.a_src, b.a_dst, b.weatt, b.alphaE, b.amax, EX);
  k_edge_exp<<<nblk(EX * 2, 256), 256, 0, stream>>>(b.dstX, b.alphaE, b.amax, b.asum, EX);
  k_edge_scatter<<<nblk(EX * (HC / 4), 256), 256, 0, stream>>>(
      b.srcX, b.dstX, b.alphaE, b.asum, b.Hfeat, b.acc, outAlpha, EX, HC);
  // 5) head mean + bias (+relu)
  k_finalize<<<nblk((long)N * Cc, 256), 256, 0, stream>>>(b.acc, bias, outFeat, N, Cc, doRelu);
}

extern "C" void kernel_launch(void* const* d_in, const int* in_sizes, int n_in,
                              void* d_out, int out_size, void* d_ws, size_t ws_size,
                              hipStream_t stream) {
  // Inputs (setup_inputs order)
  const float* x        = (const float*)d_in[0];   // [N,128]
  const int*   ei       = (const int*)d_in[1];     // [2,E]
  const float* ea       = (const float*)d_in[2];   // [E,2]
  const float* W1       = (const float*)d_in[3];
  const float* att_s1   = (const float*)d_in[4];
  const float* att_d1   = (const float*)d_in[5];
  const float* We1      = (const float*)d_in[6];
  const float* att_e1   = (const float*)d_in[7];
  const float* bias1    = (const float*)d_in[8];
  const float* W2       = (const float*)d_in[9];
  const float* att_s2   = (const float*)d_in[10];
  const float* att_d2   = (const float*)d_in[11];
  const float* We2      = (const float*)d_in[12];
  const float* att_e2   = (const float*)d_in[13];
  const float* bias2    = (const float*)d_in[14];

  const int IN_C = 128, HID = 64, HC = GAT_H * HID;   // 128
  const int N = in_sizes[0] / IN_C;                   // 100000
  const int E = in_sizes[1] / 2;                      // 1600000
  const long EX = (long)E + N;                        // with self loops

  // Workspace arena
  size_t off = 0;
  char* base = (char*)d_ws;
  auto arena = [&](size_t bytes) -> void* {
    off = (off + 255) & ~(size_t)255;
    void* p = base + off;
    off += bytes;
    return p;
  };
  float*    ssum   = (float*)arena((size_t)N * 2 * 4);
  float*    cnt    = (float*)arena((size_t)N * 4);
  int*      srcX   = (int*)arena((size_t)EX * 4);
  int*      dstX   = (int*)arena((size_t)EX * 4);
  float*    eaX    = (float*)arena((size_t)EX * 2 * 4);
  float*    Hfeat  = (float*)arena((size_t)N * HC * 4);   // reused both layers
  float*    a_src  = (float*)arena((size_t)N * 2 * 4);
  float*    a_dst  = (float*)arena((size_t)N * 2 * 4);
  unsigned* amax   = (unsigned*)arena((size_t)N * 2 * 4);
  float*    asum   = (float*)arena((size_t)N * 2 * 4);
  float*    alphaE = (float*)arena((size_t)EX * 2 * 4);
  float*    acc    = (float*)arena((size_t)N * HC * 4);   // reused both layers
  float*    h1     = (float*)arena((size_t)N * HID * 4);
  float*    weatt  = (float*)arena(4 * 4);
  (void)ws_size; (void)n_in; (void)out_size;

  // Outputs: h2 [N*64] | alpha1 [EX*2] | alpha2 [EX*2]
  float* out_h2     = (float*)d_out;
  float* out_alpha1 = out_h2 + (size_t)N * HID;
  float* out_alpha2 = out_alpha1 + (size_t)EX * 2;

  // --- self loops with mean edge attr ---
  k_zero_f32<<<nblk((long)N * 2, 256), 256, 0, stream>>>(ssum, (long)N * 2);
  k_zero_f32<<<nblk((long)N, 256), 256, 0, stream>>>(cnt, (long)N);
  k_loop_accum<<<nblk(E, 256), 256, 0, stream>>>(ei, ea, ssum, cnt, E);
  k_build_ext<<<nblk(EX, 256), 256, 0, stream>>>(ei, ea, ssum, cnt, srcX, dstX, eaX, E, N);

  LayerBufs b;
  b.srcX = srcX; b.dstX = dstX; b.eaX = eaX;
  b.Hfeat = Hfeat; b.a_src = a_src; b.a_dst = a_dst;
  b.asum = asum; b.alphaE = alphaE; b.acc = acc; b.weatt = weatt; b.amax = amax;

  // Layer 1: x[N,128] -> h1[N,64] (relu), alpha1
  run_gat_layer(x, IN_C, W1, att_s1, att_d1, We1, att_e1, bias1,
                out_alpha1, h1, /*doRelu=*/1, N, EX, b, stream);
  // Layer 2: h1[N,64] -> h2[N,64], alpha2
  run_gat_layer(h1, HID, W2, att_s2, att_d2, We2, att_e2, bias2,
                out_alpha2, out_h2, /*doRelu=*/0, N, EX, b, stream);
}